// PointerNetworks_2078764171865
// MI455X (gfx1250) — compile-verified
//
#include <hip/hip_runtime.h>
#include <hip/hip_bf16.h>

#define B_  16
#define T_  16
#define L_  512
#define H_  768

typedef __attribute__((ext_vector_type(2))) float v2f;
typedef __attribute__((ext_vector_type(8))) float v8f;

__device__ __forceinline__ float selu_f(float x) {
    const float alpha = 1.6732632423543772f;
    const float scale = 1.0507009873554805f;
    return scale * (x > 0.0f ? x : alpha * (__expf(x) - 1.0f));
}

// ---------------------------------------------------------------------------
// Row-gather index build: rowidx[b*T+t] = b*L + Yindex[b,t]
// ---------------------------------------------------------------------------
__global__ void build_rows_kernel(const int* __restrict__ Yindex,
                                  int* __restrict__ rowidx) {
    int m = threadIdx.x;              // 0..255
    int b = m / T_;
    rowidx[m] = b * L_ + Yindex[m];
}

// ---------------------------------------------------------------------------
// C[m, n] = sum_k A[row(m), k] * W[n, k]   (K = N = H_)
// One wave computes a 16x16 tile via V_WMMA_F32_16X16X4_F32, K-loop step 4.
// Fragment layouts per CDNA5 ISA 7.12.2:
//   A:   lane l -> row m0 + (l&15), elems K = k + 2*(l>>4) + {0,1}
//   B:   lane l -> col n0 + (l&15), elems K = k + 2*(l>>4) + {0,1}  (B[k][n]=W[n][k])
//   C/D: lane l, vgpr r -> C[m0 + r + 8*(l>>4)][n0 + (l&15)]
// ---------------------------------------------------------------------------
__global__ void gemm_wmma_f32_kernel(const float* __restrict__ A,
                                     const float* __restrict__ W,
                                     float* __restrict__ C,
                                     const int* __restrict__ rowidx) {
    const int lane   = threadIdx.x & 31;
    const int wave   = threadIdx.x >> 5;          // 0..7
    const int tile_m = blockIdx.x;
    const int tile_n = blockIdx.y * 8 + wave;
    const int m0 = tile_m * 16;
    const int n0 = tile_n * 16;
    const int half = lane >> 4;                   // 0 or 1
    const int sub  = lane & 15;

    const int am   = m0 + sub;
    const int arow = rowidx ? rowidx[am] : am;
    const float* aptr = A + (size_t)arow * H_ + 2 * half;
    const float* bptr = W + (size_t)(n0 + sub) * H_ + 2 * half;

    v8f c = {};
    #pragma unroll 8
    for (int k = 0; k < H_; k += 4) {
        v2f a = *(const v2f*)(aptr + k);
        v2f b = *(const v2f*)(bptr + k);
        // (neg_a, A, neg_b, B, c_mod, C, reuse_a, reuse_b)
        c = __builtin_amdgcn_wmma_f32_16x16x4_f32(false, a, false, b,
                                                  (short)0, c, false, false);
    }

    float* cptr = C + (size_t)(m0 + 8 * half) * H_ + (n0 + sub);
    #pragma unroll
    for (int r = 0; r < 8; ++r)
        cptr[(size_t)r * H_] = c[r];
}

// ---------------------------------------------------------------------------
// scores[b,t,l] = mask ? selu( V . selu(WE[b,l,:] + WD[b,t,:]) ) : -FLT_MAX
// One 128-thread block per (b,l); WE row + V in LDS; 4 waves cover 16 t's.
// ---------------------------------------------------------------------------
__global__ void scores_kernel(const float* __restrict__ WE,
                              const float* __restrict__ WD,
                              const float* __restrict__ V,
                              const int* __restrict__ Xindex,
                              const int* __restrict__ lens,
                              float* __restrict__ scores) {
    const int bl = blockIdx.x;        // 0 .. B*L-1
    const int b  = bl / L_;
    const int l  = bl % L_;

    __shared__ float sWE[H_];
    __shared__ float sV[H_];
    for (int h = threadIdx.x; h < H_; h += blockDim.x) {
        sWE[h] = WE[(size_t)bl * H_ + h];
        sV[h]  = V[h];
    }
    __syncthreads();

    const int lane = threadIdx.x & 31;
    const int wave = threadIdx.x >> 5;            // 0..3
    const int llen = lens[b];

    for (int t = wave; t < T_; t += 4) {
        const float* wd = WD + (size_t)(b * T_ + t) * H_;
        float acc = 0.0f;
        #pragma unroll 4
        for (int h = lane; h < H_; h += 32)
            acc += sV[h] * selu_f(sWE[h] + wd[h]);
        #pragma unroll
        for (int off = 16; off > 0; off >>= 1)
            acc += __shfl_xor(acc, off, 32);
        if (lane == 0) {
            float s  = selu_f(acc);
            int   x0 = Xindex[b * T_ + t];
            bool  ok = (l >= x0) && (l < llen);
            scores[(size_t)(b * T_ + t) * L_ + l] = ok ? s : -3.4028235e38f;
        }
    }
}

// ---------------------------------------------------------------------------
// Per-(b,t) masked logsumexp over L and gold NLL.  grid = 256, block = 256.
// ---------------------------------------------------------------------------
__global__ void lse_kernel(const float* __restrict__ scores,
                           const int* __restrict__ Yindex,
                           float* __restrict__ nll) {
    const int row = blockIdx.x;                   // b*T + t
    const float* s = scores + (size_t)row * L_;
    __shared__ float red[256];

    float v0 = s[threadIdx.x];
    float v1 = s[threadIdx.x + 256];

    red[threadIdx.x] = fmaxf(v0, v1);
    __syncthreads();
    for (int off = 128; off > 0; off >>= 1) {
        if (threadIdx.x < off)
            red[threadIdx.x] = fmaxf(red[threadIdx.x], red[threadIdx.x + off]);
        __syncthreads();
    }
    float mx = red[0];
    __syncthreads();

    red[threadIdx.x] = __expf(v0 - mx) + __expf(v1 - mx);
    __syncthreads();
    for (int off = 128; off > 0; off >>= 1) {
        if (threadIdx.x < off)
            red[threadIdx.x] += red[threadIdx.x + off];
        __syncthreads();
    }
    if (threadIdx.x == 0) {
        float lse  = mx + __logf(red[0]);
        float gold = s[Yindex[row]];              // inside mask by construction
        nll[row] = lse - gold;                    // -(gold - lse)
    }
}

// ---------------------------------------------------------------------------
// Final mean over 256 rows -> scalar loss.  1 block, 256 threads.
// ---------------------------------------------------------------------------
__global__ void mean_kernel(const float* __restrict__ nll,
                            float* __restrict__ out) {
    __shared__ float red[256];
    red[threadIdx.x] = nll[threadIdx.x];
    __syncthreads();
    for (int off = 128; off > 0; off >>= 1) {
        if (threadIdx.x < off)
            red[threadIdx.x] += red[threadIdx.x + off];
        __syncthreads();
    }
    if (threadIdx.x == 0)
        out[0] = red[0] * (1.0f / (B_ * T_));
}

// ---------------------------------------------------------------------------
extern "C" void kernel_launch(void* const* d_in, const int* in_sizes, int n_in,
                              void* d_out, int out_size, void* d_ws, size_t ws_size,
                              hipStream_t stream) {
    (void)in_sizes; (void)n_in; (void)out_size; (void)ws_size;

    const float* hn     = (const float*)d_in[0];   // [B,L,H]
    const float* decout = (const float*)d_in[1];   // [B,L,H]
    const float* W1     = (const float*)d_in[2];   // [H,H]
    const float* W2     = (const float*)d_in[3];   // [H,H]
    const float* V      = (const float*)d_in[4];   // [H]
    const int*   Xindex = (const int*)d_in[5];     // [B,T]
    const int*   Yindex = (const int*)d_in[6];     // [B,T]
    const int*   lens   = (const int*)d_in[7];     // [B]

    char* ws = (char*)d_ws;
    const size_t WE_BYTES = (size_t)B_ * L_ * H_ * sizeof(float);   // 25,165,824
    const size_t WD_BYTES = (size_t)B_ * T_ * H_ * sizeof(float);   //    786,432
    const size_t SC_BYTES = (size_t)B_ * T_ * L_ * sizeof(float);   //    524,288
    float* WE     = (float*)(ws);
    float* WD     = (float*)(ws + WE_BYTES);
    float* scores = (float*)(ws + WE_BYTES + WD_BYTES);
    float* nll    = (float*)(ws + WE_BYTES + WD_BYTES + SC_BYTES);
    int*   rowidx = (int*)  (ws + WE_BYTES + WD_BYTES + SC_BYTES + 1024);

    // 1. gather rows for WD
    build_rows_kernel<<<1, 256, 0, stream>>>(Yindex, rowidx);

    // 2. WE = hn . W1^T   (M = B*L = 8192)
    gemm_wmma_f32_kernel<<<dim3(B_ * L_ / 16, H_ / 128), 256, 0, stream>>>(
        hn, W1, WE, nullptr);

    // 3. WD = dj . W2^T   (M = B*T = 256, gathered rows of decoder_out)
    gemm_wmma_f32_kernel<<<dim3(B_ * T_ / 16, H_ / 128), 256, 0, stream>>>(
        decout, W2, WD, rowidx);

    // 4. fused selu/dot/selu/mask scores
    scores_kernel<<<B_ * L_, 128, 0, stream>>>(WE, WD, V, Xindex, lens, scores);

    // 5. per-row logsumexp + gold NLL
    lse_kernel<<<B_ * T_, 256, 0, stream>>>(scores, Yindex, nll);

    // 6. mean -> scalar
    mean_kernel<<<1, 256, 0, stream>>>(nll, (float*)d_out);
}